// CFMM_3779571220895
// MI455X (gfx1250) — compile-verified
//
#include <hip/hip_runtime.h>
#include <hip/hip_bf16.h>

typedef __attribute__((ext_vector_type(2))) float v2f;
typedef __attribute__((ext_vector_type(8))) float v8f;

#define DE 64  // embedding dim

// ---------------------------------------------------------------- utilities
__global__ void k_zero(float* __restrict__ p, long long n) {
  long long i = (long long)blockIdx.x * blockDim.x + threadIdx.x;
  if (i < n) p[i] = 0.0f;
}

__global__ void k_axpy(float* __restrict__ acc, const float* __restrict__ x, long long n) {
  long long i = (long long)blockIdx.x * blockDim.x + threadIdx.x;
  if (i < n) acc[i] += x[i];
}

// acc/3 -> u_g (first nU elems) and ig_base (rest)
__global__ void k_scale_split(const float* __restrict__ acc, float* __restrict__ u_out,
                              float* __restrict__ ig_base, long long nU, long long nTot) {
  long long i = (long long)blockIdx.x * blockDim.x + threadIdx.x;
  if (i >= nTot) return;
  const float inv3 = 1.0f / 3.0f;
  float v = acc[i] * inv3;
  if (i < nU) u_out[i] = v;
  else        ig_base[i - nU] = v;
}

// ------------------------------------------------------------------- SpMM
// COO scatter: out[row] += val * x[col].  One wave (32 lanes) per nonzero,
// each lane moves 2 consecutive floats of the 64-wide row.
__global__ void k_spmm(const int* __restrict__ rows, const int* __restrict__ cols,
                       const float* __restrict__ vals, const float* __restrict__ x,
                       float* __restrict__ y, int nnz) {
  long long gid = (long long)blockIdx.x * blockDim.x + threadIdx.x;
  int e    = (int)(gid >> 5);   // wave-uniform
  int lane = (int)(gid & 31);
  if (e >= nnz) return;
  if (e + 8 < nnz) {
    // warm the cache for an upcoming gather row (gfx1250 global_prefetch_b8)
    __builtin_prefetch(x + (size_t)cols[e + 8] * DE + lane * 2, 0, 1);
  }
  int r = rows[e];
  int c = cols[e];
  float v = vals[e];
  const float* xr = x + (size_t)c * DE + lane * 2;
  float*       yr = y + (size_t)r * DE + lane * 2;
  float2 xv = *(const float2*)xr;
  atomicAdd(yr,     v * xv.x);
  atomicAdd(yr + 1, v * xv.y);
}

// ------------------------------------------------------- WMMA query kernel
// scores[m] = dot(tanh(X[m,:] @ W + b), wv)   for 16 rows per wave.
// Uses V_WMMA_F32_16X16X4_F32 (full f32 precision, matches reference).
// Layouts (ISA 7.12.2, f32):
//   A 16x4 : lane(l,half) VGPR0 = (M=l, K=kb), VGPR1 = (M=l, K=kb+1), kb = 4*kt + 2*half
//   B 4x16 : lane(l,half) VGPR0 = (K=kb, N=l), VGPR1 = (K=kb+1, N=l)
//   C 16x16: lane(l,half) VGPR r = (M = r + 8*half, N = l)
__global__ void k_query_wmma(const float* __restrict__ X,   // (M,64)
                             const float* __restrict__ W,   // (64,64) row-major
                             const float* __restrict__ bias,// (64)
                             const float* __restrict__ wv,  // (64)
                             float* __restrict__ scores,    // (M)
                             int M) {
  int wave = (int)(((long long)blockIdx.x * blockDim.x + threadIdx.x) >> 5);
  int lane = threadIdx.x & 31;
  int ntiles = M >> 4;
  if (wave >= ntiles) return;          // wave-uniform: EXEC all-ones below

  int half = lane >> 4;                // 0 or 1
  int l    = lane & 15;
  int rowBase = wave << 4;

  v8f c0 = {}, c1 = {}, c2 = {}, c3 = {};
  const float* xrow = X + (size_t)(rowBase + l) * DE;

#pragma unroll
  for (int kt = 0; kt < 16; ++kt) {
    int kb = kt * 4 + half * 2;
    v2f a;
    a.x = xrow[kb];
    a.y = xrow[kb + 1];
    const float* w0 = W + (size_t)kb * DE;        // row K = kb
    const float* w1 = W + (size_t)(kb + 1) * DE;  // row K = kb+1
    v2f b0, b1, b2, b3;
    b0.x = w0[l];      b0.y = w1[l];
    b1.x = w0[l + 16]; b1.y = w1[l + 16];
    b2.x = w0[l + 32]; b2.y = w1[l + 32];
    b3.x = w0[l + 48]; b3.y = w1[l + 48];
    c0 = __builtin_amdgcn_wmma_f32_16x16x4_f32(false, a, false, b0, (short)0, c0, false, false);
    c1 = __builtin_amdgcn_wmma_f32_16x16x4_f32(false, a, false, b1, (short)0, c1, false, false);
    c2 = __builtin_amdgcn_wmma_f32_16x16x4_f32(false, a, false, b2, (short)0, c2, false, false);
    c3 = __builtin_amdgcn_wmma_f32_16x16x4_f32(false, a, false, b3, (short)0, c3, false, false);
  }

  // Fused epilogue: tanh(c + bias) * wv, then reduce over N (16 lanes/half).
  float partial[8];
#pragma unroll
  for (int r = 0; r < 8; ++r) {
    float s;
    s  = tanhf(c0[r] + bias[l])      * wv[l];
    s += tanhf(c1[r] + bias[l + 16]) * wv[l + 16];
    s += tanhf(c2[r] + bias[l + 32]) * wv[l + 32];
    s += tanhf(c3[r] + bias[l + 48]) * wv[l + 48];
    partial[r] = s;
  }
#pragma unroll
  for (int m = 1; m <= 8; m <<= 1) {
#pragma unroll
    for (int r = 0; r < 8; ++r)
      partial[r] += __shfl_xor(partial[r], m, 32);
  }
  if (l == 0) {
#pragma unroll
    for (int r = 0; r < 8; ++r)
      scores[rowBase + half * 8 + r] = partial[r];   // M = r + 8*half
  }
}

// ------------------------------------------------ softmax + combine + norm
// One wave per item row: w = softmax(qimg, qtxt); h = w0*img + w1*txt;
// h_norm = h / max(||h||, eps).
__global__ void k_fuse_h(const float* __restrict__ img, const float* __restrict__ txt,
                         const float* __restrict__ qimg, const float* __restrict__ qtxt,
                         float* __restrict__ h_out, float* __restrict__ hnorm_out, int N) {
  long long gid = (long long)blockIdx.x * blockDim.x + threadIdx.x;
  int row  = (int)(gid >> 5);
  int lane = (int)(gid & 31);
  if (row >= N) return;
  float a = qimg[row], b = qtxt[row];
  float mx = fmaxf(a, b);
  float ea = __expf(a - mx), eb = __expf(b - mx);
  float inv = 1.0f / (ea + eb);
  float w0 = ea * inv, w1 = eb * inv;
  size_t off = (size_t)row * DE + lane * 2;
  float2 iv = *(const float2*)(img + off);
  float2 tv = *(const float2*)(txt + off);
  float h0 = w0 * iv.x + w1 * tv.x;
  float h1 = w0 * iv.y + w1 * tv.y;
  float ss = h0 * h0 + h1 * h1;
#pragma unroll
  for (int m = 1; m <= 16; m <<= 1) ss += __shfl_xor(ss, m, 32);
  float nrm = 1.0f / fmaxf(sqrtf(ss), 1e-12f);
  float2 hv;  hv.x = h0;        hv.y = h1;
  float2 hn;  hn.x = h0 * nrm;  hn.y = h1 * nrm;
  *(float2*)(h_out + off)     = hv;
  *(float2*)(hnorm_out + off) = hn;
}

// Final gate: w2 = softmax(c1, c2); i_g = w2_0*ig_base + w2_1*h_norm
__global__ void k_fuse_ig(const float* __restrict__ igb, const float* __restrict__ hn,
                          const float* __restrict__ c1, const float* __restrict__ c2,
                          float* __restrict__ ig_out, long long n) {
  long long i = (long long)blockIdx.x * blockDim.x + threadIdx.x;
  if (i >= n) return;
  int row = (int)(i >> 6);
  float a = c1[row], b = c2[row];
  float mx = fmaxf(a, b);
  float ea = __expf(a - mx), eb = __expf(b - mx);
  float inv = 1.0f / (ea + eb);
  ig_out[i] = (ea * inv) * igb[i] + (eb * inv) * hn[i];
}

// ----------------------------------------------------------------- launch
static inline int cdiv_ll(long long a, long long b) { return (int)((a + b - 1) / b); }

extern "C" void kernel_launch(void* const* d_in, const int* in_sizes, int n_in,
                              void* d_out, int out_size, void* d_ws, size_t ws_size,
                              hipStream_t stream) {
  const float* user_emb = (const float*)d_in[0];
  const float* item_emb = (const float*)d_in[1];
  const float* Wq1      = (const float*)d_in[2];
  const float* bq1      = (const float*)d_in[3];
  const float* wq2      = (const float*)d_in[4];
  const float* Wc1      = (const float*)d_in[5];
  const float* bc1      = (const float*)d_in[6];
  const float* wc2      = (const float*)d_in[7];
  const float* adj_vals = (const float*)d_in[8];
  const float* img_vals = (const float*)d_in[9];
  const float* txt_vals = (const float*)d_in[10];
  const int* adj_rows   = (const int*)d_in[11];
  const int* adj_cols   = (const int*)d_in[12];
  const int* img_rows   = (const int*)d_in[13];
  const int* img_cols   = (const int*)d_in[14];
  const int* txt_rows   = (const int*)d_in[15];
  const int* txt_cols   = (const int*)d_in[16];

  const int NU = in_sizes[0] / DE;          // 80000
  const int NI = in_sizes[1] / DE;          // 30000
  const int NT = NU + NI;                   // 110000
  const int nnz_adj = in_sizes[8];          // 2,000,000
  const int nnz_mm  = in_sizes[9];          // 300,000

  const long long NUE = (long long)NU * DE;
  const long long NIE = (long long)NI * DE;
  const long long NE  = (long long)NT * DE;

  // workspace layout (floats)
  float* ws   = (float*)d_ws;
  float* bufA = ws;                 // NE  (cur / later: ig_base)
  float* bufB = ws + NE;            // NE  (nxt / later: h_norm)
  float* acc  = ws + 2 * NE;        // NE
  float* qimg = ws + 3 * NE;        // NI
  float* qtxt = qimg + NI;          // NI
  float* c1s  = qtxt + NI;          // NI
  float* c2s  = c1s + NI;           // NI

  // output layout: u_g | i_g | img_e | txt_e | h
  float* out_ug  = (float*)d_out;
  float* out_ig  = out_ug  + NUE;
  float* out_img = out_ig  + NIE;
  float* out_txt = out_img + NIE;
  float* out_h   = out_txt + NIE;

  const int B = 256;

  // ---- user-item graph propagation: ego -> 2 layers, averaged into acc/3
  hipMemcpyAsync(bufA,       user_emb, NUE * sizeof(float), hipMemcpyDeviceToDevice, stream);
  hipMemcpyAsync(bufA + NUE, item_emb, NIE * sizeof(float), hipMemcpyDeviceToDevice, stream);
  hipMemcpyAsync(acc,  bufA,           NE  * sizeof(float), hipMemcpyDeviceToDevice, stream);

  float* cur = bufA;
  float* nxt = bufB;
  for (int layer = 0; layer < 2; ++layer) {
    k_zero<<<cdiv_ll(NE, B), B, 0, stream>>>(nxt, NE);
    k_spmm<<<cdiv_ll((long long)nnz_adj * 32, B), B, 0, stream>>>(
        adj_rows, adj_cols, adj_vals, cur, nxt, nnz_adj);
    k_axpy<<<cdiv_ll(NE, B), B, 0, stream>>>(acc, nxt, NE);
    float* t = cur; cur = nxt; nxt = t;
  }
  float* ig_base = bufA;   // propagation buffers now free
  float* hnorm   = bufB;
  k_scale_split<<<cdiv_ll(NE, B), B, 0, stream>>>(acc, out_ug, ig_base, NUE, NE);

  // ---- multimodal SpMMs straight into their output slots
  k_zero<<<cdiv_ll(NIE, B), B, 0, stream>>>(out_img, NIE);
  k_zero<<<cdiv_ll(NIE, B), B, 0, stream>>>(out_txt, NIE);
  k_spmm<<<cdiv_ll((long long)nnz_mm * 32, B), B, 0, stream>>>(
      img_rows, img_cols, img_vals, item_emb, out_img, nnz_mm);
  k_spmm<<<cdiv_ll((long long)nnz_mm * 32, B), B, 0, stream>>>(
      txt_rows, txt_cols, txt_vals, item_emb, out_txt, nnz_mm);

  // ---- WMMA attention queries (16 rows per wave, 8 waves per block)
  const int ntiles = NI / 16;
  const int qgrid  = cdiv_ll(ntiles, 8);
  k_query_wmma<<<qgrid, 256, 0, stream>>>(out_img, Wq1, bq1, wq2, qimg, NI);
  k_query_wmma<<<qgrid, 256, 0, stream>>>(out_txt, Wq1, bq1, wq2, qtxt, NI);

  // ---- h = softmax-gate(img,txt), plus normalized copy
  k_fuse_h<<<cdiv_ll((long long)NI * 32, B), B, 0, stream>>>(
      out_img, out_txt, qimg, qtxt, out_h, hnorm, NI);

  // ---- cross queries on i_g and h_norm, then final gate
  k_query_wmma<<<qgrid, 256, 0, stream>>>(ig_base, Wc1, bc1, wc2, c1s, NI);
  k_query_wmma<<<qgrid, 256, 0, stream>>>(hnorm,   Wc1, bc1, wc2, c2s, NI);
  k_fuse_ig<<<cdiv_ll(NIE, B), B, 0, stream>>>(ig_base, hnorm, c1s, c2s, out_ig, NIE);
}